// DenseContrastiveLoss_67886253081209
// MI455X (gfx1250) — compile-verified
//
#include <hip/hip_runtime.h>
#include <hip/hip_bf16.h>
#include <math.h>

typedef float v2f __attribute__((ext_vector_type(2)));
typedef float v8f __attribute__((ext_vector_type(8)));
typedef __attribute__((address_space(3))) float lds_float;

#define B_   8
#define D_   128
#define S_   4096          // 64*64 positions per batch
#define TINV (1.0f / 50.0f)

// LDS byte offset of a __shared__ object (AS3 pointers are LDS offsets).
static __device__ __forceinline__ unsigned ldsAddr(float* p) {
  return (unsigned)(unsigned long long)(lds_float*)p;
}

// Async global->LDS copy, 4B per lane, GVS addressing (SGPR base + VGPR off).
static __device__ __forceinline__ void async_ld_b32(unsigned ldsA, unsigned gOff,
                                                    const float* base) {
  asm volatile("global_load_async_to_lds_b32 %0, %1, %2"
               :: "v"(ldsA), "v"(gOff), "s"(base) : "memory");
}
static __device__ __forceinline__ void wait_async0() {
  asm volatile("s_wait_asynccnt 0x0" ::: "memory");
}

// ---------------------------------------------------------------------------
// Kernel 1: inverse L2 norms of q and p per (batch, position).
// ---------------------------------------------------------------------------
__global__ __launch_bounds__(256) void norms_kernel(
    const float* __restrict__ q, const float* __restrict__ p,
    float* __restrict__ inv_nq, float* __restrict__ inv_np) {
  int gid = blockIdx.x * 256 + threadIdx.x;   // 0 .. B*S-1
  int b = gid >> 12;
  int s = gid & (S_ - 1);
  const float* qb = q + (size_t)b * D_ * S_ + s;
  const float* pb = p + (size_t)b * D_ * S_ + s;
  float sq = 0.f, sp = 0.f;
#pragma unroll 4
  for (int d = 0; d < D_; ++d) {
    float a = qb[(size_t)d * S_]; sq = fmaf(a, a, sq);
    float c = pb[(size_t)d * S_]; sp = fmaf(c, c, sp);
  }
  inv_nq[gid] = 1.0f / fmaxf(sqrtf(sq), 1e-8f);
  inv_np[gid] = 1.0f / fmaxf(sqrtf(sp), 1e-8f);
}

// ---------------------------------------------------------------------------
// Kernel 2: fused  Q·P^T (argmax of normalized sim, tracking raw dot) and
//           Q·N^T (sum exp(dot/T)) via V_WMMA_F32_16X16X4_F32, with
//           double-buffered GLOBAL_LOAD_ASYNC_TO_LDS staging.
// Grid: B * (S/128) blocks, 256 threads (8 waves); wave w owns 16 rows.
// ---------------------------------------------------------------------------
__global__ __launch_bounds__(256) void fused_kernel(
    const float* __restrict__ q, const float* __restrict__ p,
    const float* __restrict__ n,
    const float* __restrict__ inv_nq, const float* __restrict__ inv_np,
    float* __restrict__ partial) {
  // P/N tiles 128(K) x 32(N), pair-interleaved: lds[(k>>1)*64 + j*2 + (k&1)],
  // double buffered. 4 x 16KB = 64KB.
  __shared__ float ldsP[2][64 * 64];
  __shared__ float ldsN[2][64 * 64];

  const int tid  = threadIdx.x;
  const int lane = tid & 31;
  const int wave = tid >> 5;
  const int b      = blockIdx.x >> 5;
  const int rowBlk = blockIdx.x & 31;
  const int i0 = rowBlk * 128 + wave * 16;

  const size_t bo = (size_t)b * D_ * S_;
  const float* qb = q + bo;
  const float* pb = p + bo;
  const float* nb = n + bo;
  const float* invq_b = inv_nq + b * S_;
  const float* invp_b = inv_np + b * S_;

  // Per-thread staging geometry (fixed across iterations).
  const int sd = tid >> 5;                   // base depth row: d = sd + e*8
  const int sj = tid & 31;                   // column within tile

  // Issue the async copies for tile at column j0 into buffer `buf`.
  auto stage = [&](int j0, int buf) {
    unsigned lpBase = ldsAddr(&ldsP[buf][0]);
    unsigned lnBase = ldsAddr(&ldsN[buf][0]);
#pragma unroll
    for (int e = 0; e < 16; ++e) {
      int d = e * 8 + sd;                    // 0..127
      unsigned la = ((unsigned)((d >> 1) * 64 + sj * 2 + (d & 1))) * 4u;
      unsigned go = (unsigned)((d * S_ + j0 + sj) * 4);
      async_ld_b32(lpBase + la, go, pb);
      async_ld_b32(lnBase + la, go, nb);
    }
  };

  // ---- A tile (16 x 128) in WMMA 16x4 fp32 layout, kept in registers ----
  // lane 0-15: M = lane, K pair {0,1} ; lane 16-31: M = lane-16, K pair {2,3}
  const int m  = lane & 15;
  const int kb = (lane >> 4) << 1;
  v2f a[32];
#pragma unroll
  for (int k4 = 0; k4 < 32; ++k4) {
    const float* src = qb + (size_t)(4 * k4 + kb) * S_ + (i0 + m);
    a[k4].x = src[0];
    a[k4].y = src[S_];
  }

  float invq[8];
  const int mhalf = (lane >> 4) << 3;
#pragma unroll
  for (int r = 0; r < 8; ++r) invq[r] = invq_b[i0 + mhalf + r];

  float bestS[8], bestRaw[8], sumexp[8];
  int   bestIdx[8];
#pragma unroll
  for (int r = 0; r < 8; ++r) {
    bestS[r] = -1e30f; bestRaw[r] = 0.f; bestIdx[r] = 0; sumexp[r] = 0.f;
  }

  const int kbhalf = lane >> 4;

  // Prologue: stage first tile.
  stage(0, 0);
  wait_async0();
  __syncthreads();

  for (int j0 = 0; j0 < S_; j0 += 32) {
    const int buf = (j0 >> 5) & 1;
    if (j0 + 32 < S_) stage(j0 + 32, buf ^ 1);   // prefetch next tile (async)

    const float* LP = &ldsP[buf][0];
    const float* LN = &ldsN[buf][0];
#pragma unroll
    for (int jj0 = 0; jj0 < 32; jj0 += 16) {
      const int nloc = jj0 + (lane & 15);
      v8f simC = {};
      v8f negC = {};
#pragma unroll
      for (int k4 = 0; k4 < 32; ++k4) {
        const int pr = 2 * k4 + kbhalf;
        const v2f bP = *(const v2f*)&LP[pr * 64 + nloc * 2];
        const v2f bN = *(const v2f*)&LN[pr * 64 + nloc * 2];
        simC = __builtin_amdgcn_wmma_f32_16x16x4_f32(
            false, a[k4], false, bP, (short)0, simC, false, false);
        negC = __builtin_amdgcn_wmma_f32_16x16x4_f32(
            false, a[k4], false, bN, (short)0, negC, false, false);
      }
      const int jglob = j0 + jj0 + (lane & 15);
      const float invp = invp_b[jglob];
#pragma unroll
      for (int r = 0; r < 8; ++r) {
        float raw = simC[r];
        float sc  = raw * invq[r] * invp;
        if (sc > bestS[r]) {                 // ascending j => first-max kept
          bestS[r] = sc; bestRaw[r] = raw; bestIdx[r] = jglob;
        }
        sumexp[r] += __expf(negC[r] * TINV);
      }
    }
    wait_async0();       // our prefetch landed
    __syncthreads();     // everyone done reading `buf` + their prefetch landed
  }

  // ---- reduce across the 16 lanes sharing each row (wave32 half-wave) ----
  float lsum = 0.f;
#pragma unroll
  for (int r = 0; r < 8; ++r) {
    float bs = bestS[r], br = bestRaw[r], se = sumexp[r];
    int   bi = bestIdx[r];
#pragma unroll
    for (int off = 1; off < 16; off <<= 1) {
      float obs = __shfl_xor(bs, off, 32);
      float obr = __shfl_xor(br, off, 32);
      float ose = __shfl_xor(se, off, 32);
      int   obi = __shfl_xor(bi, off, 32);
      se += ose;
      if (obs > bs || (obs == bs && obi < bi)) { bs = obs; br = obr; bi = obi; }
    }
    float dpos = br * TINV;
    lsum += log1pf(se * __expf(-dpos));      // = -log(e^dp/(e^dp+sum_neg))
  }

  // Deterministic per-block sum (reuse LDS buffer 0 as scratch).
  float* sdata = &ldsP[0][0];
  if ((lane & 15) == 0) sdata[wave * 2 + (lane >> 4)] = lsum;
  __syncthreads();
  if (tid == 0) {
    float t = 0.f;
#pragma unroll
    for (int k = 0; k < 16; ++k) t += sdata[k];
    partial[blockIdx.x] = t;
  }
}

// ---------------------------------------------------------------------------
// Kernel 3: deterministic fixed-order tree over 256 partials -> mean.
// ---------------------------------------------------------------------------
__global__ __launch_bounds__(256) void finalize_kernel(
    const float* __restrict__ partial, float* __restrict__ out) {
  __shared__ float red[256];
  int t = threadIdx.x;
  red[t] = partial[t];
  __syncthreads();
  for (int s = 128; s > 0; s >>= 1) {
    if (t < s) red[t] += red[t + s];
    __syncthreads();
  }
  if (t == 0) out[0] = red[0] * (1.0f / (float)(B_ * S_));
}

extern "C" void kernel_launch(void* const* d_in, const int* in_sizes, int n_in,
                              void* d_out, int out_size, void* d_ws,
                              size_t ws_size, hipStream_t stream) {
  const float* q = (const float*)d_in[0];   // dense_img
  const float* p = (const float*)d_in[1];   // dense_pos
  const float* n = (const float*)d_in[2];   // dense_neg
  float* ws = (float*)d_ws;
  float* inv_nq  = ws;                      // B*S floats
  float* inv_np  = ws + B_ * S_;            // B*S floats
  float* partial = ws + 2 * B_ * S_;        // 256 floats

  norms_kernel<<<(B_ * S_) / 256, 256, 0, stream>>>(q, p, inv_nq, inv_np);
  fused_kernel<<<B_ * (S_ / 128), 256, 0, stream>>>(q, p, n, inv_nq, inv_np,
                                                    partial);
  finalize_kernel<<<1, 256, 0, stream>>>(partial, (float*)d_out);
}